// GCN_62921270886989
// MI455X (gfx1250) — compile-verified
//
#include <hip/hip_runtime.h>
#include <hip/hip_bf16.h>
#include <math.h>

#define N_NODES   100000
#define N_EDGES   1600000
#define IN_DIM    128
#define F_SIZE    64
#define OUT_DIM   10
#define NUM_G     512

typedef __attribute__((ext_vector_type(2))) float v2f;
typedef __attribute__((ext_vector_type(8))) float v8f;

// ---------------------------------------------------------------------------
// Dense transform Y[nrows x 64] = X[nrows x K] @ W[K x 64] using fp32 WMMA.
// One wave computes one 16x16 output tile. nrows is a multiple of 16
// (100000 = 6250*16), so EXEC is always all-ones inside the WMMA region.
//
// A layout (16x4 f32, ISA 7.12.2): lane L<16 -> M=L, VGPR0=K, VGPR1=K+1;
//                                  lane L>=16 -> M=L-16, VGPR0=K+2, VGPR1=K+3.
// B layout (4x16 f32): mirrored over N.
// C/D (16x16 f32, 8 VGPRs): VGPR r: lanes 0-15 -> M=r, lanes 16-31 -> M=r+8.
// ---------------------------------------------------------------------------
__global__ __launch_bounds__(256) void gcn_transform_wmma(
    const float* __restrict__ X, const float* __restrict__ W,
    float* __restrict__ Y, int nrows, int K) {
  const int wave = (blockIdx.x * blockDim.x + threadIdx.x) >> 5;
  const int lane = threadIdx.x & 31;
  const int rowTile = wave >> 2;   // 4 column tiles cover F_SIZE=64
  const int colTile = wave & 3;
  if (rowTile * 16 >= nrows) return;   // wave-uniform

  const int m  = lane & 15;
  const int hi = lane >> 4;            // 0 or 1
  const size_t arow = (size_t)(rowTile * 16 + m) * (size_t)K;
  const int    ncol = colTile * 16 + m;

  v8f c = {};
  for (int k = 0; k < K; k += 4) {
    const int ka = k + 2 * hi;
    v2f a, b;
    a.x = X[arow + ka];
    a.y = X[arow + ka + 1];
    b.x = W[(size_t)ka * F_SIZE + ncol];
    b.y = W[(size_t)(ka + 1) * F_SIZE + ncol];
    c = __builtin_amdgcn_wmma_f32_16x16x4_f32(
        /*neg_a=*/false, a, /*neg_b=*/false, b,
        /*c_mod=*/(short)0, c, /*reuse_a=*/false, /*reuse_b=*/false);
  }

  #pragma unroll
  for (int r = 0; r < 8; ++r) {
    const int rowOut = rowTile * 16 + r + 8 * hi;
    Y[(size_t)rowOut * F_SIZE + ncol] = c[r];
  }
}

// ---------------------------------------------------------------------------
// Degree / normalization helpers
// ---------------------------------------------------------------------------
__global__ void k_init_deg(float* __restrict__ deg) {
  int i = blockIdx.x * blockDim.x + threadIdx.x;
  if (i < N_NODES) deg[i] = 1.0f;             // self-loop contribution
}

__global__ void k_count_deg(const int* __restrict__ dst, float* __restrict__ deg) {
  int e = blockIdx.x * blockDim.x + threadIdx.x;
  if (e < N_EDGES) atomicAdd(&deg[dst[e]], 1.0f);
}

__global__ void k_finalize_dinv(float* __restrict__ deg) {
  int i = blockIdx.x * blockDim.x + threadIdx.x;
  if (i < N_NODES) deg[i] = rsqrtf(deg[i]);   // deg >= 1 always
}

__global__ void k_zero(float* __restrict__ p, int n) {
  int i = blockIdx.x * blockDim.x + threadIdx.x;
  if (i < n) p[i] = 0.0f;
}

__global__ void k_count_batch(const int* __restrict__ batch, float* __restrict__ cnt) {
  int i = blockIdx.x * blockDim.x + threadIdx.x;
  if (i < N_NODES) atomicAdd(&cnt[batch[i]], 1.0f);
}

// out[i,:] = h[i,:] * dinv[i]^2   (the self-loop message; also zero-inits out)
__global__ void k_self_init(const float* __restrict__ h, const float* __restrict__ dinv,
                            float* __restrict__ out) {
  int i = blockIdx.x * blockDim.x + threadIdx.x;
  if (i < N_NODES * F_SIZE) {
    const int node = i >> 6;
    const float w = dinv[node];
    out[i] = h[i] * w * w;
  }
}

// 16 threads per edge, 4 floats (one float4) each: out[dst] += h[src]*norm
__global__ void k_edge_scatter(const float* __restrict__ h, const float* __restrict__ dinv,
                               const int* __restrict__ src, const int* __restrict__ dst,
                               float* __restrict__ out) {
  long long tid = (long long)blockIdx.x * blockDim.x + threadIdx.x;
  int e  = (int)(tid >> 4);
  int f4 = ((int)tid & 15) * 4;
  if (e >= N_EDGES) return;
  const int s = src[e];
  const int d = dst[e];
  const float w = dinv[s] * dinv[d];
  const float4 hv = *(const float4*)(h + (size_t)s * F_SIZE + f4);
  float* o = out + (size_t)d * F_SIZE + f4;
  atomicAdd(o + 0, hv.x * w);
  atomicAdd(o + 1, hv.y * w);
  atomicAdd(o + 2, hv.z * w);
  atomicAdd(o + 3, hv.w * w);
}

// x = relu(x + b) in place, and accumulate per-graph pooled sums
__global__ void k_bias_relu_pool(float* __restrict__ x, const float* __restrict__ b,
                                 const int* __restrict__ batch,
                                 float* __restrict__ pool) {
  int i = blockIdx.x * blockDim.x + threadIdx.x;
  if (i < N_NODES * F_SIZE) {
    const int node = i >> 6;
    const int f    = i & 63;
    float v = x[i] + b[f];
    v = fmaxf(v, 0.0f);
    x[i] = v;
    atomicAdd(&pool[(size_t)batch[node] * F_SIZE + f], v);
  }
}

// ---------------------------------------------------------------------------
// Heads: g_l = (pool_l / cnt) @ Wl_l + bl_l ; z = concat(g) @ Wf + bf ; softmax
// One thread per graph (512 graphs, trivial FLOPs).
// ---------------------------------------------------------------------------
__global__ void k_head(const float* __restrict__ pool, const float* __restrict__ cnt,
                       const float* __restrict__ Wl1, const float* __restrict__ bl1,
                       const float* __restrict__ Wl2, const float* __restrict__ bl2,
                       const float* __restrict__ Wl3, const float* __restrict__ bl3,
                       const float* __restrict__ Wf,  const float* __restrict__ bf,
                       float* __restrict__ out) {
  int g = blockIdx.x * blockDim.x + threadIdx.x;
  if (g >= NUM_G) return;
  const float inv = 1.0f / fmaxf(cnt[g], 1.0f);

  const float* Wls[3] = {Wl1, Wl2, Wl3};
  const float* bls[3] = {bl1, bl2, bl3};

  float gv[3 * OUT_DIM];
  for (int l = 0; l < 3; ++l) {
    const float* p = pool + (size_t)l * NUM_G * F_SIZE + (size_t)g * F_SIZE;
    for (int j = 0; j < OUT_DIM; ++j) {
      float s = bls[l][j];
      for (int k = 0; k < F_SIZE; ++k) s += p[k] * inv * Wls[l][k * OUT_DIM + j];
      gv[l * OUT_DIM + j] = s;
    }
  }

  float z[OUT_DIM];
  float zmax = -3.4e38f;
  for (int j = 0; j < OUT_DIM; ++j) {
    float s = bf[j];
    for (int k = 0; k < 3 * OUT_DIM; ++k) s += gv[k] * Wf[k * OUT_DIM + j];
    z[j] = s;
    zmax = fmaxf(zmax, s);
  }
  float den = 0.0f;
  for (int j = 0; j < OUT_DIM; ++j) { z[j] = expf(z[j] - zmax); den += z[j]; }
  const float r = 1.0f / den;
  for (int j = 0; j < OUT_DIM; ++j) out[(size_t)g * OUT_DIM + j] = z[j] * r;
}

// ---------------------------------------------------------------------------
extern "C" void kernel_launch(void* const* d_in, const int* in_sizes, int n_in,
                              void* d_out, int out_size, void* d_ws, size_t ws_size,
                              hipStream_t stream) {
  const float* feat  = (const float*)d_in[0];
  const int*   eidx  = (const int*)  d_in[1];
  const int*   batch = (const int*)  d_in[2];
  const float* W1  = (const float*)d_in[3];
  const float* b1  = (const float*)d_in[4];
  const float* W2  = (const float*)d_in[5];
  const float* b2  = (const float*)d_in[6];
  const float* W3  = (const float*)d_in[7];
  const float* b3  = (const float*)d_in[8];
  const float* Wl1 = (const float*)d_in[9];
  const float* bl1 = (const float*)d_in[10];
  const float* Wl2 = (const float*)d_in[11];
  const float* bl2 = (const float*)d_in[12];
  const float* Wl3 = (const float*)d_in[13];
  const float* bl3 = (const float*)d_in[14];
  const float* Wf  = (const float*)d_in[15];
  const float* bf  = (const float*)d_in[16];

  const int* src = eidx;            // edge_index[0]
  const int* dst = eidx + N_EDGES;  // edge_index[1]

  // Workspace layout (floats, all 16B-aligned offsets)
  float* ws   = (float*)d_ws;
  float* dinv = ws;                                   // N
  float* h    = dinv + N_NODES;                       // N*64
  float* xbuf = h + (size_t)N_NODES * F_SIZE;         // N*64
  float* pool = xbuf + (size_t)N_NODES * F_SIZE;      // 3*512*64
  float* cnt  = pool + 3 * NUM_G * F_SIZE;            // 512  (contiguous after pool)

  const int nodeThreads = N_NODES;
  const int elemThreads = N_NODES * F_SIZE;           // 6.4M
  const long long scatThreads = (long long)N_EDGES * 16; // 25.6M
  const int poolZero = 3 * NUM_G * F_SIZE + NUM_G;

  // --- one-time setup ---
  k_zero<<<(poolZero + 255) / 256, 256, 0, stream>>>(pool, poolZero);
  k_init_deg<<<(nodeThreads + 255) / 256, 256, 0, stream>>>(dinv);
  k_count_deg<<<(N_EDGES + 255) / 256, 256, 0, stream>>>(dst, dinv);
  k_finalize_dinv<<<(nodeThreads + 255) / 256, 256, 0, stream>>>(dinv);
  k_count_batch<<<(nodeThreads + 255) / 256, 256, 0, stream>>>(batch, cnt);

  const int tfWaves  = (N_NODES / 16) * 4;            // 25000 waves
  const int tfBlocks = (tfWaves * 32 + 255) / 256;    // 3125
  const int elemBlocks = (elemThreads + 255) / 256;
  const int scatBlocks = (int)((scatThreads + 255) / 256);

  // --- layer 1: X = features (K=128) ---
  gcn_transform_wmma<<<tfBlocks, 256, 0, stream>>>(feat, W1, h, N_NODES, IN_DIM);
  k_self_init<<<elemBlocks, 256, 0, stream>>>(h, dinv, xbuf);
  k_edge_scatter<<<scatBlocks, 256, 0, stream>>>(h, dinv, src, dst, xbuf);
  k_bias_relu_pool<<<elemBlocks, 256, 0, stream>>>(xbuf, b1, batch, pool + 0 * NUM_G * F_SIZE);

  // --- layer 2 (K=64) ---
  gcn_transform_wmma<<<tfBlocks, 256, 0, stream>>>(xbuf, W2, h, N_NODES, F_SIZE);
  k_self_init<<<elemBlocks, 256, 0, stream>>>(h, dinv, xbuf);
  k_edge_scatter<<<scatBlocks, 256, 0, stream>>>(h, dinv, src, dst, xbuf);
  k_bias_relu_pool<<<elemBlocks, 256, 0, stream>>>(xbuf, b2, batch, pool + 1 * NUM_G * F_SIZE);

  // --- layer 3 (K=64) ---
  gcn_transform_wmma<<<tfBlocks, 256, 0, stream>>>(xbuf, W3, h, N_NODES, F_SIZE);
  k_self_init<<<elemBlocks, 256, 0, stream>>>(h, dinv, xbuf);
  k_edge_scatter<<<scatBlocks, 256, 0, stream>>>(h, dinv, src, dst, xbuf);
  k_bias_relu_pool<<<elemBlocks, 256, 0, stream>>>(xbuf, b3, batch, pool + 2 * NUM_G * F_SIZE);

  // --- heads + softmax ---
  k_head<<<(NUM_G + 255) / 256, 256, 0, stream>>>(
      pool, cnt, Wl1, bl1, Wl2, bl2, Wl3, bl3, Wf, bf, (float*)d_out);
}